// GraphSelfAttention_49246095016331
// MI455X (gfx1250) — compile-verified
//
#include <hip/hip_runtime.h>
#include <math.h>

typedef __attribute__((ext_vector_type(2))) float v2f;
typedef __attribute__((ext_vector_type(8))) float v8f;

#define NN 768
#define DIN 128
#define DOUT 128
#define HEADS 4
#define DH 32
#define HALF 16
#define MAXD 8

// ---------------------------------------------------------------------------
// Kernel 1: logmap0 — xt = atanh(min(||x||,1-eps)) * x / max(||x||,1e-15)
// one block (128 thr) per row
// ---------------------------------------------------------------------------
__global__ void logmap0_kernel(const float* __restrict__ x, float* __restrict__ xt) {
    __shared__ float red[DIN];
    const int base = blockIdx.x * DIN;
    const int t = threadIdx.x;
    float v = x[base + t];
    red[t] = v * v;
    __syncthreads();
    for (int s = DIN / 2; s > 0; s >>= 1) {
        if (t < s) red[t] += red[t + s];
        __syncthreads();
    }
    float nrm = fmaxf(sqrtf(red[0]), 1e-15f);
    float z = fminf(nrm, 1.0f - 1e-5f);
    xt[base + t] = atanhf(z) * v / nrm;
}

// ---------------------------------------------------------------------------
// Generic f32 WMMA GEMM, all strides/dims compile-time so every global load
// is base + constant IOFFSET (no per-iteration address math).
// One 16x16 output tile per wave via V_WMMA_F32_16X16X4_F32 (K step 4).
//   A: M x KDIM (row stride LDA), B: KDIM x 16*NT (row stride LDB)
//   C: M x 16*NT (row stride LDC), tiles = gridDim.x * waves/block
// ---------------------------------------------------------------------------
template <int LDA, int LDB, int LDC, int NT, int KDIM, int UNROLL>
__global__ void wmma_gemm_f32(const float* __restrict__ A,
                              const float* __restrict__ B,
                              const float* __restrict__ bias,
                              float* __restrict__ C) {
    const int wave = blockIdx.x * (blockDim.x >> 5) + (threadIdx.x >> 5);
    const int lane = threadIdx.x & 31;
    const int tm = wave / NT;
    const int tn = wave % NT;
    const int r  = lane & 15;
    const int hi = lane >> 4;          // 0: K pair (k0,k0+1), 1: (k0+2,k0+3)

    const float* Arow = A + (size_t)(tm * 16 + r) * LDA + hi * 2;
    const float* Bcol = B + (size_t)(hi * 2) * LDB + tn * 16 + r;

    v8f acc = {};
#pragma unroll UNROLL
    for (int k0 = 0; k0 < KDIM; k0 += 4) {
        v2f a = *reinterpret_cast<const v2f*>(Arow + k0);  // A[m][k], A[m][k+1]
        v2f b;
        b.x = Bcol[k0 * LDB];                              // B[k][n]
        b.y = Bcol[(k0 + 1) * LDB];                        // B[k+1][n]
        acc = __builtin_amdgcn_wmma_f32_16x16x4_f32(
            false, a, false, b, (short)0, acc, false, false);
    }

    const int crow = tm * 16 + hi * 8;
    const int ccol = tn * 16 + r;
    const float bv = bias ? bias[ccol] : 0.0f;
#pragma unroll
    for (int v = 0; v < 8; ++v)
        C[(size_t)(crow + v) * LDC + ccol] = acc[v] + bv;
}

// ---------------------------------------------------------------------------
// Kernel 3: per-(i,h,dist) rotated score table
// tbl[(i*4+h)*8+dist] = rot(Q_i)·rot(K_i)/sqrt(D) - slope_h*log1p(dist)
// (scores depend on j only through dist_bias[i,j] in [0,8) — 8-entry table)
// ---------------------------------------------------------------------------
__global__ void score_table_kernel(const float* __restrict__ Q,
                                   const float* __restrict__ K,
                                   float* __restrict__ tbl) {
    const int g = blockIdx.x * blockDim.x + threadIdx.x;   // 768*4*8 = 24576
    const int dist = g & 7;
    const int ih = g >> 3;
    const int h = ih & 3;
    const int i = ih >> 2;
    const float* Qp = Q + (size_t)i * DOUT + h * DH;
    const float* Kp = K + (size_t)i * DOUT + h * DH;
    const float fd = (float)dist;
    float acc = 0.0f;
#pragma unroll
    for (int d = 0; d < HALF; ++d) {
        // freq = 10000^(-d/16) = exp(-d * ln(10000)/16)
        float freq = expf(-(float)d * 0.57564627324851148f);
        float ang = fd * freq;
        float s = sinf(ang), c = cosf(ang);
        float q1 = Qp[d], q2 = Qp[d + HALF];
        float k1 = Kp[d], k2 = Kp[d + HALF];
        float qr1 = q1 * c - q2 * s, qr2 = q1 * s + q2 * c;
        float kr1 = k1 * c - k2 * s, kr2 = k1 * s + k2 * c;
        acc += qr1 * kr1 + qr2 * kr2;
    }
    const float slopes[4] = {0.25f, 0.0625f, 0.015625f, 0.00390625f};
    tbl[g] = acc * 0.17677669529663689f /* 1/sqrt(32) */
             - slopes[h] * log1pf(fd);
}

// ---------------------------------------------------------------------------
// Kernel 4: masked softmax row. One block (256 thr) per (h,i); 768 = 3*256.
// attn[h][i][j] = softmax_j( adj ? tbl[i,h,dist[i,j]] : -inf )
// ---------------------------------------------------------------------------
__global__ void softmax_kernel(const int* __restrict__ adj,
                               const int* __restrict__ dist,
                               const float* __restrict__ tbl,
                               float* __restrict__ attn) {
    const int hi = blockIdx.x;            // h*768 + i
    const int h = hi / NN;
    const int i = hi - h * NN;
    const int t = threadIdx.x;
    __shared__ float t8[MAXD];
    __shared__ float red[256];
    if (t < MAXD) t8[t] = tbl[(size_t)(i * HEADS + h) * MAXD + t];
    __syncthreads();

    const int* arow = adj + (size_t)i * NN;
    const int* drow = dist + (size_t)i * NN;
    float s[3];
    float mx = -INFINITY;
#pragma unroll
    for (int u = 0; u < 3; ++u) {
        int j = t + u * 256;
        float v = (arow[j] != 0) ? t8[drow[j]] : -INFINITY;
        s[u] = v;
        mx = fmaxf(mx, v);
    }
    red[t] = mx; __syncthreads();
    for (int w = 128; w > 0; w >>= 1) {
        if (t < w) red[t] = fmaxf(red[t], red[t + w]);
        __syncthreads();
    }
    mx = red[0]; __syncthreads();

    float e[3], sum = 0.0f;
#pragma unroll
    for (int u = 0; u < 3; ++u) { e[u] = expf(s[u] - mx); sum += e[u]; }
    red[t] = sum; __syncthreads();
    for (int w = 128; w > 0; w >>= 1) {
        if (t < w) red[t] += red[t + w];
        __syncthreads();
    }
    const float inv = 1.0f / red[0];
    float* orow = attn + (size_t)hi * NN;
#pragma unroll
    for (int u = 0; u < 3; ++u) orow[t + u * 256] = e[u] * inv;
}

// ---------------------------------------------------------------------------
// Kernel 7: expmap0 + proj, one block (128 thr) per row
// ---------------------------------------------------------------------------
__global__ void expmap_proj_kernel(const float* __restrict__ y, float* __restrict__ out) {
    __shared__ float red[DOUT];
    const int base = blockIdx.x * DOUT;
    const int t = threadIdx.x;
    float v = y[base + t];
    red[t] = v * v;
    __syncthreads();
    for (int s = DOUT / 2; s > 0; s >>= 1) {
        if (t < s) red[t] += red[t + s];
        __syncthreads();
    }
    float nrm = fmaxf(sqrtf(red[0]), 1e-15f);
    float ev = tanhf(nrm) * v / nrm;           // expmap0 (c=1)
    __syncthreads();
    red[t] = ev * ev;
    __syncthreads();
    for (int s = DOUT / 2; s > 0; s >>= 1) {
        if (t < s) red[t] += red[t + s];
        __syncthreads();
    }
    float n2 = fmaxf(sqrtf(red[0]), 1e-15f);
    const float maxnorm = 1.0f - 1e-5f;
    out[base + t] = (n2 > maxnorm) ? ev * (maxnorm / n2) : ev;
}

// ---------------------------------------------------------------------------
extern "C" void kernel_launch(void* const* d_in, const int* in_sizes, int n_in,
                              void* d_out, int out_size, void* d_ws, size_t ws_size,
                              hipStream_t stream) {
    const float* x    = (const float*)d_in[0];
    const int*   adj  = (const int*)d_in[1];
    const int*   dist = (const int*)d_in[2];
    const float* Wq   = (const float*)d_in[3];
    const float* Wk   = (const float*)d_in[4];
    const float* Wv   = (const float*)d_in[5];
    const float* Wo   = (const float*)d_in[6];
    const float* bo   = (const float*)d_in[7];

    float* out  = (float*)d_out;              // (768,128)
    float* attn = out + (size_t)NN * DOUT;    // (4,768,768)

    const size_t MAT = (size_t)NN * DOUT;     // 98304
    float* ws   = (float*)d_ws;
    float* xt   = ws;
    float* Q    = ws + MAT;
    float* Km   = ws + 2 * MAT;
    float* V    = ws + 3 * MAT;
    float* tbl  = ws + 4 * MAT;               // 768*4*8
    float* outc = tbl + (size_t)NN * HEADS * MAXD;
    float* y    = outc + MAT;

    // 1) logmap0
    logmap0_kernel<<<NN, DIN, 0, stream>>>(x, xt);

    // 2) Q/K/V GEMMs: 48x8 tiles = 384 waves -> 96 blocks of 4 waves
    //    K=128 fully unrolled: 32 WMMAs, all loads immediate-offset.
    wmma_gemm_f32<DIN, DOUT, DOUT, 8, DIN, 32><<<96, 128, 0, stream>>>(xt, Wq, nullptr, Q);
    wmma_gemm_f32<DIN, DOUT, DOUT, 8, DIN, 32><<<96, 128, 0, stream>>>(xt, Wk, nullptr, Km);
    wmma_gemm_f32<DIN, DOUT, DOUT, 8, DIN, 32><<<96, 128, 0, stream>>>(xt, Wv, nullptr, V);

    // 3) rotated-score table (768*4*8 = 24576 entries)
    score_table_kernel<<<96, 256, 0, stream>>>(Q, Km, tbl);

    // 4) masked softmax -> attn (written straight into d_out region)
    softmax_kernel<<<HEADS * NN, 256, 0, stream>>>(adj, dist, tbl, attn);

    // 5) out[i,h,d] = sum_j attn[h,i,j] * V[j,h,d] : per-head 768x768 @ 768x32
    //    48x2 tiles = 96 waves -> 24 blocks; K=768, unroll 8
    for (int h = 0; h < HEADS; ++h) {
        wmma_gemm_f32<NN, DOUT, DOUT, 2, NN, 8><<<24, 128, 0, stream>>>(
            attn + (size_t)h * NN * NN,
            V + h * DH,
            nullptr,
            outc + h * DH);
    }

    // 6) output projection with bias
    wmma_gemm_f32<DOUT, DOUT, DOUT, 8, DIN, 32><<<96, 128, 0, stream>>>(outc, Wo, bo, y);

    // 7) expmap0 + proj -> final out
    expmap_proj_kernel<<<NN, DOUT, 0, stream>>>(y, out);
}